// AlphaChebyshevProjection_56495999811645
// MI455X (gfx1250) — compile-verified
//
#include <hip/hip_runtime.h>
#include <hip/hip_bf16.h>
#include <math.h>

// Problem constants (match reference)
#define N_ACT 80      // n
#define M_CON 85      // m
#define MP    96      // m padded to 6 row-tiles of 16 (pad rows are zero)
#define PDHG_ITERS 200
#define POWER_ITERS 20

typedef __attribute__((ext_vector_type(2))) float v2f;
typedef __attribute__((ext_vector_type(8))) float v8f;
typedef __attribute__((ext_vector_type(4))) unsigned int u32x4;
typedef __attribute__((ext_vector_type(4))) int i32x4;
typedef __attribute__((ext_vector_type(8))) int i32x8;

__device__ __forceinline__ float wave_sum(float v) {
#pragma unroll
  for (int o = 16; o > 0; o >>= 1) v += __shfl_xor(v, o, 32);
  return v;
}
__device__ __forceinline__ float wave_min_f(float v) {
#pragma unroll
  for (int o = 16; o > 0; o >>= 1) v = fminf(v, __shfl_xor(v, o, 32));
  return v;
}

// ---- TDM: 1-D DMA of 6800 fp32 (27.2 KB) global -> LDS, tracked by TENSORcnt ----
// Descriptor layout per CDNA5 ISA ch.8 (CDNA5 ISA).
__device__ __forceinline__ void tdm_load_A(const float* gsrc, float* lds_dst) {
  const unsigned long long ga = (unsigned long long)(uintptr_t)gsrc;
  const unsigned lds_off = (unsigned)(uintptr_t)lds_dst;  // low 32 bits = LDS byte offset

  u32x4 g0;
  g0[0] = 1u;                                   // count=1, is_restore=0, gather=0
  g0[1] = lds_off;                              // lds_addr [63:32]
  g0[2] = (unsigned)(ga & 0xFFFFFFFFu);         // global_addr[31:0]
  g0[3] = (unsigned)((ga >> 32) & 0x1FFFFFFu)   // global_addr[56:32] -> bits[120:96]
        | (2u << 30);                           // type=2 ("image") -> bits[127:126]

  const unsigned elems = (unsigned)(M_CON * N_ACT);   // 6800 fp32
  i32x8 g1;
  g1[0] = (int)(2u << 16);                      // workgroup_mask=0, data_size=2 (4B)
  g1[1] = (int)((elems & 0xFFFFu) << 16);       // tensor_dim0[15:0] -> bits[63:48]
  g1[2] = (int)(((elems >> 16) & 0xFFFFu)       // tensor_dim0[31:16] -> bits[79:64]
        | (1u << 16));                          // tensor_dim1=1 -> bits[95:80]
  g1[3] = (int)((elems & 0xFFFFu) << 16);       // tile_dim0 -> bits[127:112]
  g1[4] = (int)1;                               // tile_dim1=1 -> bits[143:128]; tile_dim2=0
  g1[5] = (int)elems;                           // tensor_dim0_stride[31:0] -> bits[191:160]
  g1[6] = 0;                                    // stride0[47:32]=0, stride1[15:0]=0
  g1[7] = 0;                                    // tensor_dim1_stride upper = 0

  i32x4 gz = {0, 0, 0, 0};                      // groups 2/3 unused (<=2D tensor)
#if defined(__clang_major__) && (__clang_major__ >= 23)
  i32x8 gz8 = {0, 0, 0, 0, 0, 0, 0, 0};
  __builtin_amdgcn_tensor_load_to_lds(g0, g1, gz, gz, gz8, 0);
#else
  __builtin_amdgcn_tensor_load_to_lds(g0, g1, gz, gz, 0);
#endif
}

// out[0..95] = A(96x80, rows >= 85 are zero) * x[0..79]
// A fragment (16x4 f32): lane L<16: v0=A[M=L][K=0], v1=A[L][K=1]; L>=16: K=2,3.
// B fragment (4x16 f32): x replicated across the 16 N columns. D: VGPR i,
// lanes0-15 = (M=i), lanes16-31 = (M=8+i); all N columns equal, so lane0/lane16
// scatter the 16 row results to LDS.
__device__ __forceinline__ void matvec_A(const float* As, const float* x,
                                         float* out, int lane) {
  const int half = lane >> 4, l15 = lane & 15;
#pragma unroll
  for (int r = 0; r < 6; ++r) {
    v8f c = {};
#pragma unroll
    for (int k = 0; k < 20; ++k) {
      const float* ap = &As[(r * 16 + l15) * N_ACT + k * 4 + 2 * half];
      v2f a; a.x = ap[0]; a.y = ap[1];
      v2f b; b.x = x[k * 4 + 2 * half]; b.y = x[k * 4 + 2 * half + 1];
      c = __builtin_amdgcn_wmma_f32_16x16x4_f32(false, a, false, b,
                                                (short)0, c, false, false);
    }
    if (l15 == 0) {
      float* o = &out[r * 16 + half * 8];
#pragma unroll
      for (int i = 0; i < 8; ++i) o[i] = c[i];
    }
  }
}

// out[0..79] = A^T * u[0..95]   (u padded with zeros above 84)
__device__ __forceinline__ void matvec_AT(const float* As, const float* u,
                                          float* out, int lane) {
  const int half = lane >> 4, l15 = lane & 15;
#pragma unroll
  for (int r = 0; r < 5; ++r) {
    v8f c = {};
#pragma unroll
    for (int k = 0; k < 24; ++k) {
      const int i0 = k * 4 + 2 * half;
      v2f a;
      a.x = As[i0 * N_ACT + r * 16 + l15];
      a.y = As[(i0 + 1) * N_ACT + r * 16 + l15];
      v2f b; b.x = u[i0]; b.y = u[i0 + 1];
      c = __builtin_amdgcn_wmma_f32_16x16x4_f32(false, a, false, b,
                                                (short)0, c, false, false);
    }
    if (l15 == 0) {
      float* o = &out[r * 16 + half * 8];
#pragma unroll
      for (int i = 0; i < 8; ++i) o[i] = c[i];
    }
  }
}

__global__ __launch_bounds__(32)
void AlphaChebyshevProjection_kernel(const float* __restrict__ xhat,
                                     const float* __restrict__ Ag,
                                     const float* __restrict__ bg,
                                     float* __restrict__ out) {
  __shared__ float As[MP * N_ACT];                 // 30720 B, rows 85..95 zero
  __shared__ float dsv[MP], bs[M_CON];
  __shared__ float zs[MP], ys[MP], zb[MP], us[MP], gs[MP], vv[MP];

  const int p = blockIdx.x;
  const int lane = threadIdx.x;
  const float* Ap = Ag + (size_t)p * (M_CON * N_ACT);

  // ---- stage A into LDS with the Tensor Data Mover; overlap pad/b with DMA ----
  tdm_load_A(Ap, As);                                            // TENSORcnt++
  for (int t = M_CON * N_ACT + lane; t < MP * N_ACT; t += 32) As[t] = 0.f;
  for (int i = lane; i < M_CON; i += 32) bs[i] = bg[(size_t)p * M_CON + i];
  __builtin_amdgcn_s_wait_tensorcnt(0);                          // DMA complete
  __syncthreads();

  // ---- d = clip(||A_i||, 1e-12); init state ----
  for (int i = lane; i < MP; i += 32) {
    float dv = 0.f;
    if (i < M_CON) {
      float s = 0.f;
      for (int j = 0; j < N_ACT; ++j) { float a = As[i * N_ACT + j]; s += a * a; }
      dv = fmaxf(sqrtf(s), 1e-12f);
    }
    dsv[i] = dv;
    vv[i] = (i < 81) ? 1.f : 0.f;                  // v = ones(n+1)
    zs[i] = 0.f; ys[i] = 0.f; zb[i] = 0.f; us[i] = 0.f; gs[i] = 0.f;
  }
  __syncthreads();

  // ---- power iterations: v <- normalize(G^T G v) ----
  for (int it = 0; it < POWER_ITERS; ++it) {
    matvec_A(As, vv, us, lane);
    const float v80 = vv[80];
    __syncthreads();
    for (int i = lane; i < MP; i += 32) us[i] += dsv[i] * v80;   // G(v)
    __syncthreads();
    matvec_AT(As, us, gs, lane);
    float part = 0.f;
    for (int i = lane; i < M_CON; i += 32) part += dsv[i] * us[i];
    const float gr = wave_sum(part);
    __syncthreads();
    if (lane == 0) gs[80] = gr;
    __syncthreads();
    float np = 0.f;
    for (int j = lane; j < 81; j += 32) np += gs[j] * gs[j];
    const float nrm = sqrtf(wave_sum(np)) + 1e-12f;
    for (int j = lane; j < 81; j += 32) vv[j] = gs[j] / nrm;
    __syncthreads();
  }

  // ---- L = ||G(v)||, tau = sigma = 0.9/max(L,1e-6) ----
  matvec_A(As, vv, us, lane);
  const float v80f = vv[80];
  __syncthreads();
  for (int i = lane; i < MP; i += 32) us[i] += dsv[i] * v80f;
  __syncthreads();
  float lp = 0.f;
  for (int i = lane; i < M_CON; i += 32) lp += us[i] * us[i];
  const float L = sqrtf(wave_sum(lp));
  const float tau = 0.9f / fmaxf(L, 1e-6f);
  const float sigma = tau;
  __syncthreads();

  // ---- PDHG (Chambolle-Pock), 200 fixed iterations ----
  for (int it = 0; it < PDHG_ITERS; ++it) {
    // gt = G^T y  (80 WMMA cols + rank-1 d-row)
    matvec_AT(As, ys, gs, lane);
    float part = 0.f;
    for (int i = lane; i < M_CON; i += 32) part += dsv[i] * ys[i];
    const float gr = wave_sum(part);
    __syncthreads();
    if (lane == 0) gs[80] = gr;
    __syncthreads();
    // z_new = relu(z - tau*(c + gt)); z_bar = 2 z_new - z
    for (int j = lane; j < 81; j += 32) {
      const float cj = (j == 80) ? -1.f : 0.f;
      const float zn = fmaxf(zs[j] - tau * (cj + gs[j]), 0.f);
      zb[j] = 2.f * zn - zs[j];
      zs[j] = zn;
    }
    __syncthreads();
    // y = relu(y + sigma*(G(z_bar) - b))
    matvec_A(As, zb, us, lane);
    const float zb80 = zb[80];
    __syncthreads();
    for (int i = lane; i < M_CON; i += 32) {
      const float g = us[i] + dsv[i] * zb80;
      ys[i] = fmaxf(ys[i] + sigma * (g - bs[i]), 0.f);
    }
    __syncthreads();
  }

  // ---- alpha map:  out = clip(x0 + alpha*(xhat - x0), >=0) ----
  matvec_A(As, zs, us, lane);                      // A x0  (reads zs[0..79] only)
  for (int j = lane; j < N_ACT; j += 32)
    vv[j] = xhat[(size_t)p * N_ACT + j] - zs[j];   // dvec
  __syncthreads();
  matvec_A(As, vv, gs, lane);                      // A dvec
  __syncthreads();

  float amin = __builtin_inff();
  for (int i = lane; i < M_CON; i += 32) {
    const float slack = fmaxf(bs[i] - us[i], 0.f);
    const float ad = gs[i];
    const float ai = (ad > 0.f) ? (slack / (ad + 1e-12f)) : __builtin_inff();
    amin = fminf(amin, ai);
  }
  float alpha = wave_min_f(amin);
  if (!(alpha < 1e38f)) alpha = 1.f;               // where(isfinite, alpha, 1)
  alpha = fminf(fmaxf(alpha - 1e-9f, 0.f), 1.f);
  for (int j = lane; j < N_ACT; j += 32)
    out[(size_t)p * N_ACT + j] = fmaxf(zs[j] + alpha * vv[j], 0.f);
}

extern "C" void kernel_launch(void* const* d_in, const int* in_sizes, int n_in,
                              void* d_out, int out_size, void* d_ws, size_t ws_size,
                              hipStream_t stream) {
  (void)n_in; (void)out_size; (void)d_ws; (void)ws_size;
  const float* xhat = (const float*)d_in[0];   // [B,S,80] fp32
  const float* A    = (const float*)d_in[1];   // [B,S,85,80] fp32
  const float* b    = (const float*)d_in[2];   // [B,S,85] fp32
  float* out        = (float*)d_out;           // [B,S,80] fp32
  const int P = in_sizes[0] / N_ACT;           // 1024 problems
  AlphaChebyshevProjection_kernel<<<dim3(P), dim3(32), 0, stream>>>(xhat, A, b, out);
}